// NN_16_1_Pooling_58772332478911
// MI455X (gfx1250) — compile-verified
//
#include <hip/hip_runtime.h>
#include <hip/hip_bf16.h>

typedef __attribute__((ext_vector_type(2))) float v2f;
typedef __attribute__((ext_vector_type(4))) float v4f;
typedef __attribute__((ext_vector_type(8))) float v8f;

// Fixed problem geometry from the reference: x [16, 64, 224, 224] fp32, K=2.
#define DIM_W   224
#define DIM_NW  112
#define DIM_IP  56           // output-row pairs per image plane
#define DIM_BC  (16 * 64)    // batch * channels collapsed
#define N_TASKS (DIM_BC * DIM_IP)   // 57,344 row-pair tasks

// One wave per task: 2 output rows x 112 patches = 14 WMMAs
// (V_WMMA_F32_16X16X4_F32, full fp32: D(16x16) = W1(16x4)*patches(4x16)+b1).
// Width = 3 "super-groups" of 32 patches (one b128 load feeds the even-patch
// and odd-patch WMMAs) + one classic 16-patch group.
__global__ void NN_16_1_Pooling_58772332478911_kernel(
    const float* __restrict__ x,
    const float* __restrict__ W1,   // [16,4] row-major
    const float* __restrict__ b1,   // [16]
    const float* __restrict__ W2,   // [1,16]
    const float* __restrict__ b2,   // [1]
    float* __restrict__ out)        // [16,64,112,112]
{
    const int lane = threadIdx.x & 31;
    // Wave id is lane-uniform: pin it to an SGPR so the task loop, the /56
    // chain and the load/store base pointers all go scalar (SALU + SADDR
    // addressing), leaving zero per-task VALU address math.
    const int wave   = __builtin_amdgcn_readfirstlane(
                           (int)((blockIdx.x * blockDim.x + threadIdx.x) >> 5));
    const int nwaves = (int)((gridDim.x * blockDim.x) >> 5);

    const bool hi = (lane >= 16);
    const int  m  = lane & 15;

    // ---- A matrix = W1 (16x4 fp32).
    // ISA 32-bit A 16x4 layout: lanes 0-15: VGPR0=K0, VGPR1=K1 (M=lane);
    //                           lanes 16-31: VGPR0=K2, VGPR1=K3 (M=lane-16).
    const int k0 = hi ? 2 : 0;
    v2f a;
    a.x = W1[m * 4 + k0];
    a.y = W1[m * 4 + k0 + 1];

    // ---- C matrix = b1 broadcast; D layout: VGPR r, lane l ->
    //      (M = r + 8*(l>=16), N = l%16). Layer-2 weights per lane,
    //      b2 pre-folded (each half contributes b2/2).
    v8f c;
    float w2v[8];
    #pragma unroll
    for (int r = 0; r < 8; ++r) {
        c[r]   = b1[r + (hi ? 8 : 0)];
        w2v[r] = W2[r + (hi ? 8 : 0)];
    }
    const float halfb2 = 0.5f * b2[0];

    // Loop-invariant per-lane offsets (elements).
    const int xoff4 = (hi ? DIM_W : 0) + 4 * m;          // b128 super-group loads
    const int xoff2 = (hi ? DIM_W : 0) + 2 * m + 192;    // leftover b64 load
    const int ooff  = (hi ? DIM_NW : 0) + 2 * m;         // paired b64 stores
    const int ooffL = (hi ? DIM_NW : 0) + m + 96;        // leftover b32 store

    for (int task = wave; task < N_TASKS; task += nwaves) {
        const int bc = task / DIM_IP;          // image plane (SALU)
        const int ip = task - bc * DIM_IP;     // output-row pair (SALU)

        // Scalar base pointers; lane offsets + immediates do the rest.
        const float* xs = x   + (size_t)(bc * DIM_W  + 4 * ip) * DIM_W;
        float*       os = out + (size_t)(bc * DIM_NW + 2 * ip) * DIM_NW;

        v2f   resS[2][3];   // {even,odd} patch-pair results per super-group
        float resL[2];      // leftover 16-patch group

        #pragma unroll
        for (int rr = 0; rr < 2; ++rr) {          // output rows 2*ip, 2*ip+1
            #pragma unroll
            for (int s = 0; s < 3; ++s) {         // 32-patch super-groups
                // b128: cols 64s+4m..+3 of this lane's input row.
                // Lo lanes carry K0,K1 (top row); hi lanes K2,K3 (bottom).
                const v4f F = *(const v4f*)(xs + xoff4 + rr * (2 * DIM_W) + 64 * s);
                v2f bE; bE.x = F.x; bE.y = F.y;   // patch 32s + 2m
                v2f bO; bO.x = F.z; bO.y = F.w;   // patch 32s + 2m + 1

                const v8f dE = __builtin_amdgcn_wmma_f32_16x16x4_f32(
                    false, a, false, bE, (short)0, c, false, false);
                const v8f dO = __builtin_amdgcn_wmma_f32_16x16x4_f32(
                    false, a, false, bO, (short)0, c, false, false);

                float pE = halfb2, pO = halfb2;
                #pragma unroll
                for (int r = 0; r < 8; ++r) {
                    const float hE = dE[r] > 0.0f ? dE[r] : 0.0f;
                    const float hO = dO[r] > 0.0f ? dO[r] : 0.0f;
                    pE = fmaf(w2v[r], hE, pE);
                    pO = fmaf(w2v[r], hO, pO);
                }
                // Combine complementary hidden halves; afterwards every lane
                // holds the full result for its patch pair.
                resS[rr][s].x = pE + __shfl_xor(pE, 16, 32);
                resS[rr][s].y = pO + __shfl_xor(pO, 16, 32);
            }

            // Leftover group: patches 96..111 (cols 192..223), b64 load.
            const v2f L = *(const v2f*)(xs + xoff2 + rr * (2 * DIM_W));
            const v8f dL = __builtin_amdgcn_wmma_f32_16x16x4_f32(
                false, a, false, L, (short)0, c, false, false);
            float pL = halfb2;
            #pragma unroll
            for (int r = 0; r < 8; ++r) {
                const float hL = dL[r] > 0.0f ? dL[r] : 0.0f;
                pL = fmaf(w2v[r], hL, pL);
            }
            resL[rr] = pL + __shfl_xor(pL, 16, 32);
        }

        // Branch-free full-wave stores: one b64 per super-group covers 32
        // patches of row 2*ip (lo lanes) + 32 of row 2*ip+1 (hi lanes).
        #pragma unroll
        for (int s = 0; s < 3; ++s) {
            *(v2f*)(os + ooff + 32 * s) = hi ? resS[1][s] : resS[0][s];
        }
        os[ooffL] = hi ? resL[1] : resL[0];
    }
}

extern "C" void kernel_launch(void* const* d_in, const int* in_sizes, int n_in,
                              void* d_out, int out_size, void* d_ws, size_t ws_size,
                              hipStream_t stream) {
    (void)in_sizes; (void)n_in; (void)out_size; (void)d_ws; (void)ws_size;
    const float* x  = (const float*)d_in[0];
    const float* W1 = (const float*)d_in[1];
    const float* b1 = (const float*)d_in[2];
    const float* W2 = (const float*)d_in[3];
    const float* b2 = (const float*)d_in[4];
    float* out = (float*)d_out;

    // 57,344 tasks / (1792 blocks * 8 waves) = exactly 4 tasks per wave.
    dim3 block(256);   // 8 wave32 waves per block
    dim3 grid(1792);
    NN_16_1_Pooling_58772332478911_kernel<<<grid, block, 0, stream>>>(
        x, W1, b1, W2, b2, out);
}